// DeformableConvBlock_50663434224349
// MI455X (gfx1250) — compile-verified
//
#include <hip/hip_runtime.h>

// Problem constants (from reference)
#define B_     8
#define CIN_   64
#define COUT_  64
#define H_     96
#define W_     96
#define HW_    (H_ * W_)          // 9216
#define KTAPS_ 9                  // 3x3
#define KDIM_  (CIN_ * KTAPS_)    // 576
#define NPIX_  (B_ * HW_)         // 73728
#define EPS_   1e-5f

typedef float v2f __attribute__((ext_vector_type(2)));
typedef float v8f __attribute__((ext_vector_type(8)));

union fiu { float f; int i; };

// ---------------------------------------------------------------------------
// Kernel 0: zero the BN accumulators (sums[64] + sumsq[64] + stats[128])
// ---------------------------------------------------------------------------
__global__ void k_zero(float* __restrict__ p) {
    p[threadIdx.x] = 0.0f;
}

// ---------------------------------------------------------------------------
// Kernel 1: offset conv (18 ch, 3x3) + bake bilinear sampling metadata.
// One thread per pixel. w_off (18*576 floats = 41.5 KB) staged in LDS.
// Metadata per (pixel, tap): 4 clamped flat indices + 4 masked weights.
// ---------------------------------------------------------------------------
__global__ __launch_bounds__(256) void k_offset_meta(
    const float* __restrict__ x, const float* __restrict__ w_off,
    const float* __restrict__ b_off, float* __restrict__ meta)
{
    __shared__ float wl[18 * KDIM_];               // 41472 B
    for (int e = threadIdx.x; e < 18 * KDIM_; e += 256) wl[e] = w_off[e];
    __syncthreads();

    const int pix = blockIdx.x * 256 + threadIdx.x;    // 288 blocks x 256
    const int b   = pix / HW_;
    const int hw  = pix - b * HW_;
    const int h   = hw / W_;
    const int w   = hw - h * W_;

    float acc[18];
    #pragma unroll
    for (int j = 0; j < 18; ++j) acc[j] = b_off[j];

    const float* xb = x + (size_t)b * CIN_ * HW_;
    for (int c = 0; c < CIN_; ++c) {
        float xv[9];
        #pragma unroll
        for (int q = 0; q < 9; ++q) {
            const int yy = h + q / 3 - 1;
            const int xx = w + q % 3 - 1;
            const bool ok = (yy >= 0) & (yy < H_) & (xx >= 0) & (xx < W_);
            xv[q] = ok ? xb[c * HW_ + yy * W_ + xx] : 0.0f;
        }
        #pragma unroll
        for (int j = 0; j < 18; ++j) {
            const float* wr = &wl[j * KDIM_ + c * 9];
            #pragma unroll
            for (int q = 0; q < 9; ++q) acc[j] = fmaf(xv[q], wr[q], acc[j]);
        }
    }

    float* mp = meta + (size_t)pix * 72;           // 9 taps * 8 words
    #pragma unroll
    for (int k = 0; k < KTAPS_; ++k) {
        const float dy = acc[2 * k];
        const float dx = acc[2 * k + 1];
        const float py = (float)h + (float)(k / 3 - 1) + dy;
        const float px = (float)w + (float)(k % 3 - 1) + dx;
        const float y0f = floorf(py), x0f = floorf(px);
        const float ly = py - y0f, lx = px - x0f;
        const int y0 = (int)y0f, x0 = (int)x0f;
        const int y1 = y0 + 1,   x1 = x0 + 1;
        const bool vy0 = (y0 >= 0) & (y0 < H_), vy1 = (y1 >= 0) & (y1 < H_);
        const bool vx0 = (x0 >= 0) & (x0 < W_), vx1 = (x1 >= 0) & (x1 < W_);
        const int cy0 = y0 < 0 ? 0 : (y0 > H_ - 1 ? H_ - 1 : y0);
        const int cy1 = y1 < 0 ? 0 : (y1 > H_ - 1 ? H_ - 1 : y1);
        const int cx0 = x0 < 0 ? 0 : (x0 > W_ - 1 ? W_ - 1 : x0);
        const int cx1 = x1 < 0 ? 0 : (x1 > W_ - 1 ? W_ - 1 : x1);
        fiu i00, i01, i10, i11;
        i00.i = cy0 * W_ + cx0;  i01.i = cy0 * W_ + cx1;
        i10.i = cy1 * W_ + cx0;  i11.i = cy1 * W_ + cx1;
        float* m = mp + k * 8;
        m[0] = i00.f; m[1] = i01.f; m[2] = i10.f; m[3] = i11.f;
        m[4] = (vy0 & vx0) ? (1.f - ly) * (1.f - lx) : 0.f;
        m[5] = (vy0 & vx1) ? (1.f - ly) * lx         : 0.f;
        m[6] = (vy1 & vx0) ? ly * (1.f - lx)         : 0.f;
        m[7] = (vy1 & vx1) ? ly * lx                 : 0.f;
    }
}

// ---------------------------------------------------------------------------
// Kernel 2: fused deformable-sampling + GEMM via V_WMMA_F32_16X16X4_F32.
// Block = 256 threads = 8 waves; wave tile = 16 pixels x 64 out-channels.
// LDS: metadata (128 pix * 9 taps * 8 w = 36 KB) + w_def pre-swizzled into
// B fragments (144 kk * 4 ntiles * 32 lanes * 2 f32 = 144 KB) -> 180 KB of
// the 320 KB WGP LDS (CDNA5-only capacity).
// ---------------------------------------------------------------------------
__global__ __launch_bounds__(256) void k_dconv_wmma(
    const float* __restrict__ x, const float* __restrict__ wdef,
    const float* __restrict__ bdef, const float* __restrict__ meta,
    float* __restrict__ raw, float* __restrict__ sums, float* __restrict__ sumsq)
{
    extern __shared__ float smem[];
    float* sMeta = smem;                  // 128*72 = 9216 floats
    float* sB    = smem + 128 * 72;       // 144*4*32*2 = 36864 floats

    const int tid  = threadIdx.x;
    const int lane = tid & 31;
    const int wave = tid >> 5;
    const int pixBase = blockIdx.x * 128;             // 576 blocks

    // ---- stage metadata for this block's 128 pixels (b128 copies) ----
    {
        const uint4* src = (const uint4*)(meta + (size_t)pixBase * 72);
        uint4* dst = (uint4*)sMeta;
        for (int e = tid; e < (128 * 72) / 4; e += 256) dst[e] = src[e];
    }
    // ---- stage w_def into B-fragment layout:
    // entry(kk,n,lane) holds B[K=4kk+klo+j][N=16n+lane%16], klo=(lane>=16)?2:0
    for (int e = tid; e < 144 * 4 * 32; e += 256) {
        const int kk = e >> 7;
        const int r  = e & 127;
        const int ln = r & 31;
        const int n  = r >> 5;
        const int K0 = 4 * kk + ((ln >> 4) << 1);
        const int o  = 16 * n + (ln & 15);
        sB[e * 2 + 0] = wdef[o * KDIM_ + K0 + 0];
        sB[e * 2 + 1] = wdef[o * KDIM_ + K0 + 1];
    }
    __syncthreads();

    const int wavePix = pixBase + wave * 16;          // 16 contiguous pixels
    const int b       = wavePix / HW_;                // tile never straddles b
    const int hwBase  = wavePix - b * HW_;
    const float* xb   = x + (size_t)b * CIN_ * HW_;
    const int mp      = (wave * 16 + (lane & 15)) * 72;   // this lane's pixel
    const int klo     = (lane >> 4) << 1;

    v8f acc[4];
    #pragma unroll
    for (int n = 0; n < 4; ++n) acc[n] = (v8f){0,0,0,0,0,0,0,0};

    #pragma unroll 4
    for (int kk = 0; kk < 144; ++kk) {
        // Build fp32 A fragment (16x4 tile): lane holds K = 4kk+klo+{0,1}
        v2f a;
        #pragma unroll
        for (int j = 0; j < 2; ++j) {
            const int K  = 4 * kk + klo + j;
            const int c  = K / 9;
            const int kq = K - 9 * c;
            const float* m = &sMeta[mp + kq * 8];
            fiu u0, u1, u2, u3;
            u0.f = m[0]; u1.f = m[1]; u2.f = m[2]; u3.f = m[3];
            const float* pl = xb + c * HW_;
            if (j == 0 && c + 1 < CIN_)
                __builtin_prefetch(pl + HW_, 0, 0);   // global_prefetch_b8
            a[j] = m[4] * pl[u0.i] + m[5] * pl[u1.i]
                 + m[6] * pl[u2.i] + m[7] * pl[u3.i];
        }
        const float* bb = &sB[(kk * 128 + lane) * 2];
        const v2f b0 = *(const v2f*)(bb + 0);
        const v2f b1 = *(const v2f*)(bb + 64);
        const v2f b2 = *(const v2f*)(bb + 128);
        const v2f b3 = *(const v2f*)(bb + 192);
        acc[0] = __builtin_amdgcn_wmma_f32_16x16x4_f32(false, a, false, b0, (short)0, acc[0], false, false);
        acc[1] = __builtin_amdgcn_wmma_f32_16x16x4_f32(false, a, false, b1, (short)0, acc[1], false, false);
        acc[2] = __builtin_amdgcn_wmma_f32_16x16x4_f32(false, a, false, b2, (short)0, acc[2], false, false);
        acc[3] = __builtin_amdgcn_wmma_f32_16x16x4_f32(false, a, false, b3, (short)0, acc[3], false, false);
    }

    // ---- epilogue: +bias, store raw NCHW, accumulate BN sums ----
    #pragma unroll
    for (int n = 0; n < 4; ++n) {
        const int ch = 16 * n + (lane & 15);
        const float bias = bdef[ch];
        float s = 0.f, sq = 0.f;
        float* rp = raw + (size_t)(b * COUT_ + ch) * HW_ + hwBase;
        #pragma unroll
        for (int r = 0; r < 8; ++r) {
            const int M = (lane < 16) ? r : (r + 8);   // C/D VGPR layout
            const float v = acc[n][r] + bias;
            rp[M] = v;
            s += v; sq += v * v;
        }
        atomicAdd(&sums[ch],  s);
        atomicAdd(&sumsq[ch], sq);
    }
}

// ---------------------------------------------------------------------------
// Kernel 3: finalize per-channel mean / rstd
// ---------------------------------------------------------------------------
__global__ void k_stats(const float* __restrict__ sums,
                        const float* __restrict__ sumsq,
                        float* __restrict__ stats)
{
    const int ch = threadIdx.x;
    if (ch < COUT_) {
        const float inv  = 1.0f / (float)NPIX_;
        const float mean = sums[ch] * inv;
        const float var  = sumsq[ch] * inv - mean * mean;
        stats[ch]         = mean;
        stats[64 + ch]    = rsqrtf(var + EPS_);
    }
}

// ---------------------------------------------------------------------------
// Kernel 4: fused BN affine + ReLU
// ---------------------------------------------------------------------------
__global__ __launch_bounds__(256) void k_bn_relu(
    const float* __restrict__ raw, const float* __restrict__ stats,
    const float* __restrict__ gamma, const float* __restrict__ beta,
    float* __restrict__ out, int n)
{
    int i = blockIdx.x * blockDim.x + threadIdx.x;
    const int stride = gridDim.x * blockDim.x;
    for (; i < n; i += stride) {
        const int ch = (i / HW_) & (COUT_ - 1);
        float v = (raw[i] - stats[ch]) * stats[64 + ch];
        v = v * gamma[ch] + beta[ch];
        out[i] = v > 0.f ? v : 0.f;
    }
}

// ---------------------------------------------------------------------------
extern "C" void kernel_launch(void* const* d_in, const int* in_sizes, int n_in,
                              void* d_out, int out_size, void* d_ws, size_t ws_size,
                              hipStream_t stream) {
    const float* x     = (const float*)d_in[0];
    const float* w_off = (const float*)d_in[1];
    const float* b_off = (const float*)d_in[2];
    const float* w_def = (const float*)d_in[3];
    const float* b_def = (const float*)d_in[4];
    const float* gamma = (const float*)d_in[5];
    const float* beta  = (const float*)d_in[6];
    float* out = (float*)d_out;

    // workspace layout
    float* meta  = (float*)d_ws;                       // NPIX*72 floats (21.2 MB)
    float* raw   = meta + (size_t)NPIX_ * 72;          // NPIX*64 floats (18.9 MB)
    float* sums  = raw + (size_t)NPIX_ * COUT_;        // 64
    float* sumsq = sums + 64;                          // 64
    float* stats = sumsq + 64;                         // 128 (mean | rstd)

    k_zero<<<1, 256, 0, stream>>>(sums);
    k_offset_meta<<<NPIX_ / 256, 256, 0, stream>>>(x, w_off, b_off, meta);

    const size_t lds = (size_t)(128 * 72 + 144 * 4 * 32 * 2) * sizeof(float); // 184320 B
    k_dconv_wmma<<<NPIX_ / 128, 256, lds, stream>>>(x, w_def, b_def, meta,
                                                    raw, sums, sumsq);
    k_stats<<<1, 64, 0, stream>>>(sums, sumsq, stats);

    const int n = NPIX_ * COUT_;
    k_bn_relu<<<2048, 256, 0, stream>>>(raw, stats, gamma, beta, out, n);
}